// BVPVelocityLoss_55061480735175
// MI455X (gfx1250) — compile-verified
//
#include <hip/hip_runtime.h>
#include <hip/hip_bf16.h>
#include <float.h>

// MI455X / gfx1250, wave32. FFT-bound loss: WMMA-accelerated radix-16 Stockham
// FFT in 256KB LDS (320KB/WGP), one workgroup (256 thr = 8 waves) per row.

typedef __attribute__((ext_vector_type(2))) float v2f;   // 16x4 f32 A / 4x16 f32 B fragment (2 VGPR/lane)
typedef __attribute__((ext_vector_type(8))) float v8f;   // 16x16 f32 C/D fragment (8 VGPR/lane)

#define NFFT   16384
#define NHALF  8193          // rfft bin count N//2+1
#define SEQ    16384
#define NROWS  512
#define NTHR   256
#define K16    1024          // NFFT/16 : constant k-stride for all radix-16 stages (s*m == N/16)
#define TWO_PI 6.28318530717958647692f

// accumulator slots at the head of d_ws
#define ACC_PEAR 0
#define ACC_COS  1
#define ACC_PABS 2
#define ACC_PT   3
#define ACC_NMI  4
#define ACC_N    16

// ---------------------------------------------------------------------------
// Precomputed A-fragments for the 16x16 DFT matrix W16[m][k] = exp(dir*i*2pi*m*k/16).
// Layout per ISA 7.12.2 (32-bit A 16x4): lanes 0-15 hold row M=lane with K=(0,1)
// in the two VGPRs; lanes 16-31 hold K=(2,3). Independent of stage params.
// ---------------------------------------------------------------------------
struct W16Frag { v2f wr[4], wi[4], win[4]; };

__device__ __forceinline__ W16Frag make_w16(float dir, int tid)
{
    const int lane = tid & 31;
    const int half = lane >> 4;
    const int ln   = lane & 15;
    W16Frag W;
#pragma unroll
    for (int kc = 0; kc < 4; ++kc) {
        const int   k0 = 4 * kc + 2 * half;
        const float a0 = (TWO_PI / 16.0f) * (float)(ln * k0);
        const float a1 = (TWO_PI / 16.0f) * (float)(ln * (k0 + 1));
        v2f wr = { __cosf(a0),       __cosf(a1)       };
        v2f wi = { dir * __sinf(a0), dir * __sinf(a1) };
        v2f wn = { -wi.x, -wi.y };
        W.wr[kc] = wr; W.wi[kc] = wi; W.win[kc] = wn;
    }
    return W;
}

// ---------------------------------------------------------------------------
// One Stockham radix-16 stage:  y[q + s*(16p + j)] = twiddle(j,p) * sum_k x[q + s*(p + m*k)] * W16^{jk}
// 16 columns at a time as a 16x16 complex matmul on V_WMMA_F32_16X16X4_F32
// (K split into 4 chained WMMAs; complex = 4 real matmuls -> 16 WMMA/tile).
// Twiddles generated by complex recurrence (2 sin/cos pairs per tile instead of 16).
// ---------------------------------------------------------------------------
__device__ __forceinline__ void radix16_stage_wmma(
    const float* sr, const float* si, float* dr, float* di,
    int n, int ls /*log2(s)*/, float dir, const W16Frag& W, int tid)
{
    const int   s    = 1 << ls;
    const float th0  = TWO_PI / (float)n;
    const int   lane = tid & 31;
    const int   wave = tid >> 5;     // 0..7
    const int   half = lane >> 4;
    const int   ln   = lane & 15;

    // total columns = m*s = N/16 = 1024 -> 64 tiles of 16 columns, 8 per wave (uniform: EXEC all 1s)
    for (int T = wave; T < 64; T += 8) {
        const int col  = T * 16 + ln;
        const int p    = col >> ls;
        const int q    = col & (s - 1);
        const int base = q + (p << ls);

        v2f bXr[4], bXi[4];
#pragma unroll
        for (int kc = 0; kc < 4; ++kc) {
            const int k0 = 4 * kc + 2 * half;
            const int i0 = base + K16 * k0;
            const int i1 = i0 + K16;
            v2f xr = { sr[i0], sr[i1] };
            v2f xi = { si[i0], si[i1] };
            bXr[kc] = xr; bXi[kc] = xi;
        }

        v8f accR = {};  // Yr = Wr*Xr - Wi*Xi
        v8f accI = {};  // Yi = Wi*Xr + Wr*Xi
#pragma unroll
        for (int kc = 0; kc < 4; ++kc)
            accR = __builtin_amdgcn_wmma_f32_16x16x4_f32(false, W.wr[kc],  false, bXr[kc], (short)0, accR, false, false);
#pragma unroll
        for (int kc = 0; kc < 4; ++kc)
            accR = __builtin_amdgcn_wmma_f32_16x16x4_f32(false, W.win[kc], false, bXi[kc], (short)0, accR, false, false);
#pragma unroll
        for (int kc = 0; kc < 4; ++kc)
            accI = __builtin_amdgcn_wmma_f32_16x16x4_f32(false, W.wi[kc],  false, bXr[kc], (short)0, accI, false, false);
#pragma unroll
        for (int kc = 0; kc < 4; ++kc)
            accI = __builtin_amdgcn_wmma_f32_16x16x4_f32(false, W.wr[kc],  false, bXi[kc], (short)0, accI, false, false);

        // twiddle + scatter: D VGPR v holds row j = v + 8*half of this lane's column.
        // w(j) = exp(dir*i*th0*p*j) generated by recurrence from w(8*half) and w1 = w(1).
        const float fp = (float)p;
        const float a1 = th0 * fp;
        const float c1 = __cosf(a1), s1 = dir * __sinf(a1);
        const float a0 = a1 * (float)(8 * half);
        float cw = __cosf(a0), sw = dir * __sinf(a0);
        const int obase = q + ((p << 4) << ls);
#pragma unroll
        for (int v = 0; v < 8; ++v) {
            const int   j  = v + 8 * half;
            const float vr = accR[v], vi = accI[v];
            const int   o  = obase + (j << ls);
            dr[o] = vr * cw - vi * sw;
            di[o] = vr * sw + vi * cw;
            const float ncw = cw * c1 - sw * s1;   // w *= w1 (dir folded into s-terms)
            const float nsw = cw * s1 + sw * c1;
            cw = ncw; sw = nsw;
        }
    }
}

// final Stockham stage: n=4, s=4096, p=0 (twiddle-free radix-4)
__device__ __forceinline__ void radix4_stage(
    const float* sr, const float* si, float* dr, float* di, float dir, int tid)
{
    for (int q = tid; q < 4096; q += NTHR) {
        float u0r = sr[q],         u0i = si[q];
        float u1r = sr[q + 4096],  u1i = si[q + 4096];
        float u2r = sr[q + 8192],  u2i = si[q + 8192];
        float u3r = sr[q + 12288], u3i = si[q + 12288];
        float t0r = u0r + u2r, t0i = u0i + u2i;
        float t1r = u0r - u2r, t1i = u0i - u2i;
        float t2r = u1r + u3r, t2i = u1i + u3i;
        float t3r = u1r - u3r, t3i = u1i - u3i;
        float w3r = -dir * t3i, w3i = dir * t3r;   // dir*i*(u1-u3)
        dr[q]         = t0r + t2r;  di[q]         = t0i + t2i;
        dr[q + 4096]  = t1r + w3r;  di[q + 4096]  = t1i + w3i;
        dr[q + 8192]  = t0r - t2r;  di[q + 8192]  = t0i - t2i;
        dr[q + 12288] = t1r - w3r;  di[q + 12288] = t1i - w3i;
    }
}

// full 16384-point complex FFT; in/out in (Ar,Ai); (Br,Bi) is ping-pong scratch
__device__ __forceinline__ void fft16384(float* Ar, float* Ai, float* Br, float* Bi, float dir, int tid)
{
    const W16Frag W = make_w16(dir, tid);   // stage-invariant DFT-16 A-fragments
    __syncthreads();
    radix16_stage_wmma(Ar, Ai, Br, Bi, 16384, 0, dir, W, tid); __syncthreads();
    radix16_stage_wmma(Br, Bi, Ar, Ai, 1024,  4, dir, W, tid); __syncthreads();
    radix16_stage_wmma(Ar, Ai, Br, Bi, 64,    8, dir, W, tid); __syncthreads();
    radix4_stage      (Br, Bi, Ar, Ai, dir, tid);              __syncthreads();
}

// ---------------------------------------------------------------------------
// Kernel 1: per-row Pearson (raw sums; Pearson is invariant to the global affine
// standardization in the reference, so result is mathematically identical).
// ---------------------------------------------------------------------------
__global__ __launch_bounds__(NTHR) void pearson_kernel(
    const float* __restrict__ preds, const float* __restrict__ targs,
    const int* __restrict__ pi, float* __restrict__ acc)
{
    const int r   = blockIdx.x;
    const int tid = threadIdx.x;
    const float* x = preds + (size_t)(*pi) * (size_t)(NROWS * SEQ) + (size_t)r * SEQ;
    const float* t = targs + (size_t)r * SEQ;

    float sx = 0.f, sy = 0.f, sxy = 0.f, sx2 = 0.f, sy2 = 0.f;
    for (int k = tid; k < SEQ; k += NTHR) {
        __builtin_prefetch(&x[k + 8 * NTHR], 0, 1);   // global_prefetch_b8
        float a = x[k], b = t[k];
        sx += a; sy += b; sxy += a * b; sx2 += a * a; sy2 += b * b;
    }
    __shared__ float red[5 * NTHR];
    red[tid] = sx; red[NTHR + tid] = sy; red[2 * NTHR + tid] = sxy;
    red[3 * NTHR + tid] = sx2; red[4 * NTHR + tid] = sy2;
    for (int off = NTHR / 2; off > 0; off >>= 1) {
        __syncthreads();
        if (tid < off)
            for (int g = 0; g < 5; ++g)
                red[g * NTHR + tid] += red[g * NTHR + tid + off];
    }
    __syncthreads();
    if (tid == 0) {
        const float N   = (float)SEQ;
        const float num = N * red[2 * NTHR] - red[0] * red[NTHR];
        const float den = sqrtf((N * red[3 * NTHR] - red[0] * red[0]) *
                                (N * red[4 * NTHR] - red[NTHR] * red[NTHR]));
        atomicAdd(&acc[ACC_PEAR], 1.0f - num / den);
    }
}

// ---------------------------------------------------------------------------
// Kernel 2: per-row spectral work (phase correlation + power spectrum)
// ---------------------------------------------------------------------------
__global__ __launch_bounds__(NTHR) void fft_row_kernel(
    const float* __restrict__ preds, const float* __restrict__ targs,
    const int* __restrict__ pi, const int* __restrict__ pepoch,
    float* __restrict__ acc, v2f* __restrict__ gXw)
{
    if (*pepoch < 400) return;
    const int r   = blockIdx.x;
    const int tid = threadIdx.x;
    const float* x = preds + (size_t)(*pi) * (size_t)(NROWS * SEQ) + (size_t)r * SEQ;
    const float* t = targs + (size_t)r * SEQ;

    __shared__ float Ar[NFFT], Ai[NFFT], Br[NFFT], Bi[NFFT];   // 256 KB ping-pong
    __shared__ float pxx[NHALF];                               // |rfft(x)|^2 bins
    __shared__ float red[2 * NTHR];
    __shared__ int   redi[NTHR];

    // 1) Xw = FFT(x * hann)  -> stash full spectrum in d_ws (streaming, NT hint)
    for (int k = tid; k < NFFT; k += NTHR) {
        float w = 0.5f * (1.0f - __cosf(TWO_PI * (float)k / (float)NFFT));
        Ar[k] = x[k] * w; Ai[k] = 0.f;
    }
    fft16384(Ar, Ai, Br, Bi, -1.0f, tid);
    v2f* gx = gXw + (size_t)r * (size_t)NFFT;
    for (int k = tid; k < NFFT; k += NTHR) {
        v2f v = { Ar[k], Ai[k] };
        __builtin_nontemporal_store(v, &gx[k]);
    }
    __syncthreads();

    // 2) Tw = FFT(t * hann)
    for (int k = tid; k < NFFT; k += NTHR) {
        float w = 0.5f * (1.0f - __cosf(TWO_PI * (float)k / (float)NFFT));
        Ar[k] = t[k] * w; Ai[k] = 0.f;
    }
    fft16384(Ar, Ai, Br, Bi, -1.0f, tid);

    // 3) phase-only correlation, then inverse FFT; argmax of real part
    for (int k = tid; k < NFFT; k += NTHR) {
        v2f xw = __builtin_nontemporal_load(&gx[k]);
        float tr = Ar[k], ti = Ai[k];
        float cr = xw.x * tr + xw.y * ti;      // Xw * conj(Tw)
        float ci = xw.y * tr - xw.x * ti;
        float inv = __frsqrt_rn(cr * cr + ci * ci + 1e-30f);
        Ar[k] = cr * inv; Ai[k] = ci * inv;
    }
    fft16384(Ar, Ai, Br, Bi, +1.0f, tid);  // (1/N scale irrelevant for argmax)
    float bv = -FLT_MAX; int bk = 0;
    for (int k = tid; k < NFFT; k += NTHR) {
        float v = Ar[k];
        if (v > bv || (v == bv && k < bk)) { bv = v; bk = k; }
    }
    red[tid] = bv; redi[tid] = bk;
    for (int off = NTHR / 2; off > 0; off >>= 1) {
        __syncthreads();
        if (tid < off) {
            float v2 = red[tid + off]; int k2 = redi[tid + off];
            if (v2 > red[tid] || (v2 == red[tid] && k2 < redi[tid])) { red[tid] = v2; redi[tid] = k2; }
        }
    }
    __syncthreads();
    if (tid == 0) atomicAdd(&acc[ACC_COS], __cosf(TWO_PI * (float)redi[0] / (float)NFFT));
    __syncthreads();

    // 4) Pxx = |FFT(x)|^2 on rfft bins
    for (int k = tid; k < NFFT; k += NTHR) { Ar[k] = x[k]; Ai[k] = 0.f; }
    fft16384(Ar, Ai, Br, Bi, -1.0f, tid);
    for (int k = tid; k < NHALF; k += NTHR) pxx[k] = Ar[k] * Ar[k] + Ai[k] * Ai[k];
    __syncthreads();

    // 5) Ptt, accumulate sum|Pxx-Ptt| and sum Ptt  (mean(|d|)/mean(t) == sums ratio)
    for (int k = tid; k < NFFT; k += NTHR) { Ar[k] = t[k]; Ai[k] = 0.f; }
    fft16384(Ar, Ai, Br, Bi, -1.0f, tid);
    float sa = 0.f, st = 0.f;
    for (int k = tid; k < NHALF; k += NTHR) {
        float d = Ar[k] * Ar[k] + Ai[k] * Ai[k];
        sa += fabsf(pxx[k] - d); st += d;
    }
    red[tid] = sa; red[NTHR + tid] = st;
    for (int off = NTHR / 2; off > 0; off >>= 1) {
        __syncthreads();
        if (tid < off) { red[tid] += red[tid + off]; red[NTHR + tid] += red[NTHR + tid + off]; }
    }
    __syncthreads();
    if (tid == 0) { atomicAdd(&acc[ACC_PABS], red[0]); atomicAdd(&acc[ACC_PT], red[NTHR]); }
}

// ---------------------------------------------------------------------------
// Kernel 3: per-row normalized mutual information (10x10 histogram in LDS)
// ---------------------------------------------------------------------------
__global__ __launch_bounds__(NTHR) void mi_kernel(
    const float* __restrict__ preds, const float* __restrict__ targs,
    const int* __restrict__ pi, const int* __restrict__ pepoch,
    float* __restrict__ acc)
{
    if (*pepoch < 700) return;
    const int r   = blockIdx.x;
    const int tid = threadIdx.x;
    const float* x = preds + (size_t)(*pi) * (size_t)(NROWS * SEQ) + (size_t)r * SEQ;
    const float* y = targs + (size_t)r * SEQ;

    __shared__ float hist[100];
    __shared__ float red[4 * NTHR];
    __shared__ float sXmn, sXmx, sYmn, sYmx;

    float xmn = FLT_MAX, xmx = -FLT_MAX, ymn = FLT_MAX, ymx = -FLT_MAX;
    for (int k = tid; k < SEQ; k += NTHR) {
        float a = x[k], b = y[k];
        xmn = fminf(xmn, a); xmx = fmaxf(xmx, a);
        ymn = fminf(ymn, b); ymx = fmaxf(ymx, b);
    }
    red[tid] = xmn; red[NTHR + tid] = xmx; red[2 * NTHR + tid] = ymn; red[3 * NTHR + tid] = ymx;
    for (int off = NTHR / 2; off > 0; off >>= 1) {
        __syncthreads();
        if (tid < off) {
            red[tid]            = fminf(red[tid],            red[tid + off]);
            red[NTHR + tid]     = fmaxf(red[NTHR + tid],     red[NTHR + tid + off]);
            red[2 * NTHR + tid] = fminf(red[2 * NTHR + tid], red[2 * NTHR + tid + off]);
            red[3 * NTHR + tid] = fmaxf(red[3 * NTHR + tid], red[3 * NTHR + tid + off]);
        }
    }
    __syncthreads();
    if (tid == 0) { sXmn = red[0]; sXmx = red[NTHR]; sYmn = red[2 * NTHR]; sYmx = red[3 * NTHR]; }
    if (tid < 100) hist[tid] = 0.f;
    __syncthreads();

    const float ibx = 10.0f / (sXmx - sXmn);
    const float iby = 10.0f / (sYmx - sYmn);
    for (int k = tid; k < SEQ; k += NTHR) {
        int ix = (int)((x[k] - sXmn) * ibx); ix = ix < 0 ? 0 : (ix > 9 ? 9 : ix);
        int iy = (int)((y[k] - sYmn) * iby); iy = iy < 0 ? 0 : (iy > 9 ? 9 : iy);
        atomicAdd(&hist[ix * 10 + iy], 1.0f);   // ds_add_f32
    }
    __syncthreads();

    if (tid == 0) {
        const float denom = (float)NROWS * (float)SEQ;   // faithful: batch*seq
        const float eps   = 1e-8f;
        float hx[10], hy[10];
        for (int i = 0; i < 10; ++i) { hx[i] = 0.f; hy[i] = 0.f; }
        for (int i = 0; i < 10; ++i)
            for (int j = 0; j < 10; ++j) { hx[i] += hist[i * 10 + j]; hy[j] += hist[i * 10 + j]; }
        float mi = 0.f;
        for (int i = 0; i < 10; ++i)
            for (int j = 0; j < 10; ++j) {
                float pxy = hist[i * 10 + j] / denom;
                float px  = hx[i] / denom, py = hy[j] / denom;
                mi += pxy * __logf((pxy + eps) / (px * py + eps));
            }
        float hxe = 0.f, hye = 0.f;
        for (int i = 0; i < 10; ++i) {
            float px = hx[i] / denom, py = hy[i] / denom;
            hxe -= px * __logf(px + eps);
            hye -= py * __logf(py + eps);
        }
        atomicAdd(&acc[ACC_NMI], mi / ((hxe + hye) * 0.5f));
    }
}

// ---------------------------------------------------------------------------
__global__ void zero_acc_kernel(float* acc)
{
    if (threadIdx.x < ACC_N) acc[threadIdx.x] = 0.f;
}

__global__ void final_kernel(const int* pepoch, const float* acc, float* out)
{
    const int epoch = *pepoch;
    float loss = acc[ACC_PEAR] / (float)NROWS;
    if (epoch >= 400) {
        loss += 1.0f - acc[ACC_COS] / (float)NROWS;
        loss += acc[ACC_PABS] / acc[ACC_PT];
    }
    if (epoch >= 700) loss += 1.0f - acc[ACC_NMI] / (float)NROWS;
    out[0] = loss;
}

// ---------------------------------------------------------------------------
extern "C" void kernel_launch(void* const* d_in, const int* in_sizes, int n_in,
                              void* d_out, int out_size, void* d_ws, size_t ws_size,
                              hipStream_t stream)
{
    (void)in_sizes; (void)n_in; (void)out_size; (void)ws_size;
    const float* preds  = (const float*)d_in[0];   // [2, 512, 16384] f32
    const float* targs  = (const float*)d_in[1];   // [512, 16384] f32
    const int*   pi     = (const int*)d_in[2];
    const int*   pepoch = (const int*)d_in[3];

    float* acc = (float*)d_ws;                     // 16 accumulators
    v2f*   gXw = (v2f*)((float*)d_ws + ACC_N);     // 512 * 16384 complex spectrum stash

    zero_acc_kernel<<<1, 64, 0, stream>>>(acc);
    pearson_kernel<<<NROWS, NTHR, 0, stream>>>(preds, targs, pi, acc);
    fft_row_kernel<<<NROWS, NTHR, 0, stream>>>(preds, targs, pi, pepoch, acc, gXw);
    mi_kernel<<<NROWS, NTHR, 0, stream>>>(preds, targs, pi, pepoch, acc);
    final_kernel<<<1, 1, 0, stream>>>(pepoch, acc, (float*)d_out);
}